// NNCorr_90675349553999
// MI455X (gfx1250) — compile-verified
//
#include <hip/hip_runtime.h>
#include <hip/hip_bf16.h>
#include <math.h>

typedef __attribute__((ext_vector_type(2))) float v2f;
typedef __attribute__((ext_vector_type(4))) float v4f;
typedef __attribute__((ext_vector_type(8))) float v8f;

#define B_  4
#define N1_ 4096
#define N2_ 4096
#define D_  128
#define BT  64      // output tile edge (M=n2 and N=n1)
#define LDP 68      // padded LDS row stride in floats (BT + 4) -> conflict-free column gathers

// ---------------------------------------------------------------------------
// Tiled cdist: corr[b, m0+i, n0+j] = || x2[b,m0+i,:] - x1[b,n0+j,:] ||_2
// via sq2 + sq1 - 2 * (x2 . x1) with the inner products on V_WMMA_F32_16X16X4_F32.
// 256 threads = 8 waves; waves tile the 64x64 block as 4 (M) x 2 (N),
// each wave = 16x32 = two 16x16 f32 accumulators.
// ---------------------------------------------------------------------------
__global__ __launch_bounds__(256)
void cdist_tile_wmma(const float* __restrict__ x1,
                     const float* __restrict__ x2,
                     float* __restrict__ corr) {
    __shared__ float As[BT * LDP];   // x2 tile (M rows)
    __shared__ float Bs[BT * LDP];   // x1 tile (N rows)
    __shared__ float sq2[BT];
    __shared__ float sq1[BT];

    const int b  = blockIdx.z;
    const int m0 = blockIdx.y * BT;          // n2 base
    const int n0 = blockIdx.x * BT;          // n1 base
    const int t  = threadIdx.x;

    const float* x2p = x2 + ((size_t)b * N2_ + m0) * D_;
    const float* x1p = x1 + ((size_t)b * N1_ + n0) * D_;

    // Cooperative tile load: 64 rows x 128 f32 = 2048 float4 per tensor, 8 per thread.
    for (int i = 0; i < 8; ++i) {
        int f   = t + i * 256;               // float4 index within tile
        int row = f >> 5;                    // 32 float4 per row
        int c4  = (f & 31) << 2;             // float offset in row
        v4f a  = *(const v4f*)(x2p + row * D_ + c4);
        v4f bb = *(const v4f*)(x1p + row * D_ + c4);
        *(v4f*)(&As[row * LDP + c4]) = a;
        *(v4f*)(&Bs[row * LDP + c4]) = bb;
    }
    __syncthreads();

    // Row squared-norms from LDS (threads 0..63 -> sq2, 64..127 -> sq1).
    if (t < BT) {
        float s = 0.f;
        for (int d = 0; d < D_; ++d) { float v = As[t * LDP + d]; s += v * v; }
        sq2[t] = s;
    } else if (t < 2 * BT) {
        int r = t - BT;
        float s = 0.f;
        for (int d = 0; d < D_; ++d) { float v = Bs[r * LDP + d]; s += v * v; }
        sq1[r] = s;
    }
    __syncthreads();

    const int wave = t >> 5;                 // 0..7
    const int lane = t & 31;
    const int half = lane >> 4;              // 0: K pair {0,1}; 1: K pair {2,3} (ISA A 16x4 layout)
    const int mr   = lane & 15;

    const int wr = wave & 3;                 // M sub-tile index (4 x 16 rows)
    const int wc = wave >> 2;                // N half index    (2 x 32 cols)

    v8f acc0 = {};
    v8f acc1 = {};

    const int aoff  = (wr * 16 + mr)      * LDP + 2 * half;   // A frag base
    const int boff0 = (wc * 32 + mr)      * LDP + 2 * half;   // B frag (N tile 0)
    const int boff1 = (wc * 32 + 16 + mr) * LDP + 2 * half;   // B frag (N tile 1)

    for (int k0 = 0; k0 < D_; k0 += 4) {
        v2f a  = *(const v2f*)(&As[aoff  + k0]);
        v2f b0 = *(const v2f*)(&Bs[boff0 + k0]);
        v2f b1 = *(const v2f*)(&Bs[boff1 + k0]);
        acc0 = __builtin_amdgcn_wmma_f32_16x16x4_f32(false, a, false, b0,
                                                     (short)0, acc0, false, false);
        acc1 = __builtin_amdgcn_wmma_f32_16x16x4_f32(false, a, false, b1,
                                                     (short)0, acc1, false, false);
    }

    // Epilogue: D VGPR r holds (M=r, N=lane) for lanes 0-15 and (M=r+8, N=lane-16) for 16-31.
    const int nl0 = wc * 32 + mr;
    float* cb = corr + (size_t)b * N2_ * N1_;
    for (int r = 0; r < 8; ++r) {
        int   mloc = wr * 16 + r + 8 * half;
        float s2   = sq2[mloc];
        size_t base = (size_t)(m0 + mloc) * N1_ + n0;
        float d2a = s2 + sq1[nl0]      - 2.f * acc0[r];
        float d2b = s2 + sq1[nl0 + 16] - 2.f * acc1[r];
        cb[base + nl0]      = sqrtf(fmaxf(d2a, 0.f));
        cb[base + nl0 + 16] = sqrtf(fmaxf(d2b, 0.f));
    }
}

// ---------------------------------------------------------------------------
// max_ind21: argmin over N1 for each (b, n2) row. One block per row.
// ---------------------------------------------------------------------------
__global__ __launch_bounds__(256)
void argmin_rows(const float* __restrict__ corr, float* __restrict__ ind21) {
    __shared__ float sv[256];
    __shared__ int   si[256];
    const int row = blockIdx.x;              // flat b*N2 + n2
    const float* p = corr + (size_t)row * N1_;
    const int t = threadIdx.x;

    float best = 3.402823466e38f;
    int   bi   = N1_;
    for (int j = t; j < N1_; j += 256) {     // per-thread indices ascend -> first-min kept
        float v = p[j];
        if (v < best) { best = v; bi = j; }
    }
    sv[t] = best; si[t] = bi;
    __syncthreads();
    for (int s = 128; s > 0; s >>= 1) {
        if (t < s) {
            float v = sv[t + s]; int i2 = si[t + s];
            if (v < sv[t] || (v == sv[t] && i2 < si[t])) { sv[t] = v; si[t] = i2; }
        }
        __syncthreads();
    }
    if (t == 0) ind21[row] = (float)si[0];
}

// ---------------------------------------------------------------------------
// max_ind12: argmin over N2 for each (b, n1) column. One thread per column,
// coalesced row sweeps; ascending r keeps first occurrence on ties.
// ---------------------------------------------------------------------------
__global__ __launch_bounds__(256)
void argmin_cols(const float* __restrict__ corr, float* __restrict__ ind12) {
    const int b   = blockIdx.y;
    const int n1c = blockIdx.x * 256 + threadIdx.x;
    const float* p = corr + (size_t)b * N2_ * N1_ + n1c;

    float best = 3.402823466e38f;
    int   bi   = 0;
    for (int r = 0; r < N2_; ++r) {
        float v = p[(size_t)r * N1_];
        if (v < best) { best = v; bi = r; }
    }
    ind12[(size_t)b * N1_ + n1c] = (float)bi;
}

// ---------------------------------------------------------------------------
// Outputs concatenated flat: x1 | x2 | corr_mat | max_ind12 | max_ind21
// ---------------------------------------------------------------------------
extern "C" void kernel_launch(void* const* d_in, const int* in_sizes, int n_in,
                              void* d_out, int out_size, void* d_ws, size_t ws_size,
                              hipStream_t stream) {
    (void)in_sizes; (void)n_in; (void)out_size; (void)d_ws; (void)ws_size;

    const float* x1 = (const float*)d_in[0];
    const float* x2 = (const float*)d_in[1];
    float* out = (float*)d_out;

    const size_t x1n = (size_t)B_ * N1_ * D_;
    const size_t x2n = (size_t)B_ * N2_ * D_;
    const size_t cn  = (size_t)B_ * N2_ * N1_;

    float* o_x1 = out;
    float* o_x2 = out + x1n;
    float* corr = out + x1n + x2n;
    float* i12  = corr + cn;                 // [B, N1]
    float* i21  = i12 + (size_t)B_ * N1_;    // [B, N2]

    hipMemcpyAsync(o_x1, x1, x1n * sizeof(float), hipMemcpyDeviceToDevice, stream);
    hipMemcpyAsync(o_x2, x2, x2n * sizeof(float), hipMemcpyDeviceToDevice, stream);

    dim3 grid(N1_ / BT, N2_ / BT, B_);
    cdist_tile_wmma<<<grid, 256, 0, stream>>>(x1, x2, corr);
    argmin_rows<<<dim3(B_ * N2_), 256, 0, stream>>>(corr, i21);
    argmin_cols<<<dim3(N1_ / 256, B_), 256, 0, stream>>>(corr, i12);
}